// ProductQuantization_67121748902070
// MI455X (gfx1250) — compile-verified
//
#include <hip/hip_runtime.h>

typedef __attribute__((ext_vector_type(2))) float v2f;
typedef __attribute__((ext_vector_type(8))) float v8f;

#define D_DIM   256
#define K_SUB   16
#define B_CENT  256
#define D_SUB   16
#define ROWP    20                      // padded row stride (floats): 80B rows -> 16B align, conflict-free
#define BUF_FLT (B_CENT * ROWP)         // 5120 floats = 20KB per buffer
#define C2_FLT  (2 * BUF_FLT)           // c2 array starts after the two buffers

__device__ __forceinline__ void async_cb_row_b128(unsigned int lds_off, const float* gsrc) {
    // per-lane 16B: copies from the global address in gsrc to the LDS byte address in lds_off; tracked by ASYNCcnt
    asm volatile("global_load_async_to_lds_b128 %0, %1, off"
                 :: "v"(lds_off), "v"(gsrc) : "memory");
}

__global__ __launch_bounds__(256)
void ProductQuantization_pq_wmma_kernel(const float* __restrict__ vectors,
                                        const float* __restrict__ codebook,
                                        float* __restrict__ out_idx,    // V*K (index as float)
                                        float* __restrict__ out_recon,  // V*D
                                        int V)
{
    __shared__ __attribute__((aligned(16))) float smem[2 * BUF_FLT + B_CENT];

    const int tid  = threadIdx.x;
    const int lane = tid & 31;
    const int wave = tid >> 5;
    const int m    = lane & 15;   // A: row M / B: col N / C: col N
    const int half = lane >> 4;   // selects K pair (A/B) and M+8 (C)
    const int off  = half * 2;

    const int vwave = blockIdx.x * 128 + wave * 16;

    // ---- prologue: async-stage codebook[0] into buffer 0 (4 x b128 per lane) ----
    {
        const float* src = codebook + (size_t)tid * D_SUB;       // k = 0
        unsigned int dst = (unsigned int)tid * (ROWP * 4);       // byte offset, 16B aligned
        async_cb_row_b128(dst +  0, src + 0);
        async_cb_row_b128(dst + 16, src + 4);
        async_cb_row_b128(dst + 32, src + 8);
        async_cb_row_b128(dst + 48, src + 12);
    }

    for (int k = 0; k < K_SUB; ++k) {
        const int   bufo = (k & 1) * BUF_FLT;     // float offset of current buffer
        const float* buf = smem + bufo;

        // ---- issue async stage of codebook[k+1] into the other buffer, then wait
        //      for codebook[k] (async loads complete in order) ----
        if (k + 1 < K_SUB) {
            const float* src = codebook + ((size_t)(k + 1) * B_CENT + tid) * D_SUB;
            unsigned int dst = (unsigned int)(((k + 1) & 1) * (BUF_FLT * 4))
                             + (unsigned int)tid * (ROWP * 4);
            async_cb_row_b128(dst +  0, src + 0);
            async_cb_row_b128(dst + 16, src + 4);
            async_cb_row_b128(dst + 32, src + 8);
            async_cb_row_b128(dst + 48, src + 12);
            asm volatile("s_wait_asynccnt 0x4" ::: "memory");    // k's 4 transfers done
        } else {
            asm volatile("s_wait_asynccnt 0x0" ::: "memory");
        }
        __syncthreads();                                         // all waves' transfers visible

        // ---- per-centroid ||c||^2 from LDS ----
        {
            const float* row = buf + tid * ROWP;
            float s = 0.0f;
            #pragma unroll
            for (int j = 0; j < D_SUB; ++j) s = fmaf(row[j], row[j], s);
            smem[C2_FLT + tid] = s;
        }
        __syncthreads();

        // ---- load A tile: 16 vectors x 16 dims, fp32 16x4 WMMA layout (streamed, NT) ----
        int vld = vwave + m;
        if (vld >= V) vld = V - 1;           // clamp (keeps EXEC all-1s for WMMA)
        const float* xrow = vectors + (size_t)vld * D_DIM + k * D_SUB;
        v2f a0 = __builtin_nontemporal_load((const v2f*)(xrow + 0  + off));
        v2f a1 = __builtin_nontemporal_load((const v2f*)(xrow + 4  + off));
        v2f a2 = __builtin_nontemporal_load((const v2f*)(xrow + 8  + off));
        v2f a3 = __builtin_nontemporal_load((const v2f*)(xrow + 12 + off));

        float bestV[8];
        int   bestI[8];
        #pragma unroll
        for (int r = 0; r < 8; ++r) { bestV[r] = 3.4e38f; bestI[r] = 0; }

        // ---- 16 centroid groups of 16; xc via 4 chained f32 WMMAs each.
        //      Unrolled so WMMAs of tile t+1 overlap the VALU argmin of tile t. ----
        #pragma unroll 4
        for (int t = 0; t < B_CENT / 16; ++t) {
            const float* crow = buf + (t * 16 + m) * ROWP;
            v2f b0 = *(const v2f*)(crow + 0  + off);
            v2f b1 = *(const v2f*)(crow + 4  + off);
            v2f b2 = *(const v2f*)(crow + 8  + off);
            v2f b3 = *(const v2f*)(crow + 12 + off);

            v8f acc = {};
            acc = __builtin_amdgcn_wmma_f32_16x16x4_f32(false, a0, false, b0, (short)0, acc, false, false);
            acc = __builtin_amdgcn_wmma_f32_16x16x4_f32(false, a1, false, b1, (short)0, acc, false, false);
            acc = __builtin_amdgcn_wmma_f32_16x16x4_f32(false, a2, false, b2, (short)0, acc, false, false);
            acc = __builtin_amdgcn_wmma_f32_16x16x4_f32(false, a3, false, b3, (short)0, acc, false, false);

            // rank(d2) == rank(c2 - 2*xc): drop ||x||^2
            const float c2v = smem[C2_FLT + t * 16 + m];
            const int   idx = t * 16 + m;
            #pragma unroll
            for (int r = 0; r < 8; ++r) {
                float d = fmaf(-2.0f, acc[r], c2v);
                if (d < bestV[r]) { bestV[r] = d; bestI[r] = idx; }  // strict <: first index wins
            }
        }

        // ---- branchless argmin reduction across the 16 lanes of each half (N axis) ----
        #pragma unroll
        for (int w = 1; w < 16; w <<= 1) {
            #pragma unroll
            for (int r = 0; r < 8; ++r) {
                float ov = __shfl_xor(bestV[r], w, 32);
                int   oi = __shfl_xor(bestI[r], w, 32);
                bool take = (ov < bestV[r]) | ((ov == bestV[r]) & (oi < bestI[r]));
                bestV[r] = take ? ov : bestV[r];
                bestI[r] = take ? oi : bestI[r];
            }
        }

        // ---- write indices + reconstruction (lanes of a half cooperate; NT stores) ----
        #pragma unroll
        for (int r = 0; r < 8; ++r) {
            int v = vwave + r + 8 * half;      // C layout: component r -> M = r + 8*half
            if (v < V) {
                int bi = bestI[r];             // uniform across the half after reduction
                __builtin_nontemporal_store(buf[bi * ROWP + m],
                                            out_recon + (size_t)v * D_DIM + k * D_SUB + m);
                if (m == 0)
                    __builtin_nontemporal_store((float)bi, out_idx + (size_t)v * K_SUB + k);
            }
        }

        __syncthreads();   // all reads of buf done before next iter's async overwrites it
    }
}

extern "C" void kernel_launch(void* const* d_in, const int* in_sizes, int n_in,
                              void* d_out, int out_size, void* d_ws, size_t ws_size,
                              hipStream_t stream) {
    (void)n_in; (void)out_size; (void)d_ws; (void)ws_size;
    const float* vectors  = (const float*)d_in[0];   // (V, 256) f32
    const float* codebook = (const float*)d_in[1];   // (16, 256, 16) f32
    const int V = in_sizes[0] / D_DIM;

    float* out_idx   = (float*)d_out;                       // V*K
    float* out_recon = (float*)d_out + (size_t)V * K_SUB;   // V*D

    const int blocks = (V + 127) / 128;   // 128 vectors per 256-thread block (8 waves x 16)
    ProductQuantization_pq_wmma_kernel<<<blocks, 256, 0, stream>>>(
        vectors, codebook, out_idx, out_recon, V);
}